// EdgeScoreGNN_28810640622034
// MI455X (gfx1250) — compile-verified
//
#include <hip/hip_runtime.h>
#include <hip/hip_bf16.h>
#include <stdint.h>

// ---------------------------------------------------------------------------
// 2-layer GCN: out = GCN2(relu(GCN1(x))) on MI455X (gfx1250, wave32).
// Dense transform uses V_WMMA_F32_16X16X4_F32 (fp32-exact, matrix-core path).
// Edge aggregation uses L2-resident f32 atomics; norm precomputed once.
// ---------------------------------------------------------------------------

typedef __attribute__((ext_vector_type(2))) float v2f;
typedef __attribute__((ext_vector_type(8))) float v8f;

#define IND 128   // input feature dim
#define HID 16    // hidden dim

__global__ void zero_f32(float* __restrict__ p, int n) {
  int i = blockIdx.x * blockDim.x + threadIdx.x;
  if (i < n) p[i] = 0.0f;
}

// deg[d] += 1 for every edge destination (self-loop added later as +1)
__global__ void deg_count(const long long* __restrict__ dst, int E,
                          float* __restrict__ deg) {
  int e = blockIdx.x * blockDim.x + threadIdx.x;
  if (e < E) atomicAdd(&deg[(int)dst[e]], 1.0f);
}

// deg -> deg_inv_sqrt in place (self-loop => deg+1 >= 1, rsqrt always valid)
__global__ void deg_to_dis(float* __restrict__ deg, int N) {
  int i = blockIdx.x * blockDim.x + threadIdx.x;
  if (i < N) deg[i] = rsqrtf(deg[i] + 1.0f);
}

// norm[e] = dis[src] * dis[dst]   (shared by both layers)
__global__ void edge_norm(const long long* __restrict__ src,
                          const long long* __restrict__ dst,
                          const float* __restrict__ dis,
                          float* __restrict__ norm, int E) {
  int e = blockIdx.x * blockDim.x + threadIdx.x;
  if (e < E) norm[e] = dis[(int)src[e]] * dis[(int)dst[e]];
}

// h[N,16] = x[N,128] @ W1^T[128,16] using V_WMMA_F32_16X16X4_F32.
// One wave = one 16-row tile. A: lane m=lane&15, K-pair 2*(lane>>4).
// B[k][n] = W1[n][k]: lane n=lane&15, same K-pair -> contiguous float2 loads.
__global__ void gemm1_wmma(const float* __restrict__ x,
                           const float* __restrict__ W1,
                           float* __restrict__ h, int N) {
  const int lane = threadIdx.x & 31;
  const int wave = threadIdx.x >> 5;
  const int tile = blockIdx.x * (blockDim.x >> 5) + wave;
  const int row0 = tile * 16;
  if (row0 + 16 > N) return;            // whole-wave exit, EXEC stays all-1s

  const int m    = lane & 15;
  const int half = lane >> 4;           // 0: K={0,1}, 1: K={2,3}
  const float* xrow = x  + (size_t)(row0 + m) * IND + 2 * half;
  const float* wrow = W1 + (size_t)m * IND + 2 * half;   // B column m of W1^T

  v8f c = {0.f, 0.f, 0.f, 0.f, 0.f, 0.f, 0.f, 0.f};
#pragma unroll
  for (int k = 0; k < IND; k += 4) {
    v2f a = *(const v2f*)(xrow + k);
    v2f b = *(const v2f*)(wrow + k);
    // (neg_a, A, neg_b, B, c_mod, C, reuse_a, reuse_b)
    c = __builtin_amdgcn_wmma_f32_16x16x4_f32(false, a, false, b,
                                              (short)0, c, false, false);
  }

  // D layout: col = lane&15, rows 8*half + r in VGPR r -> coalesced stores
  const int col = lane & 15;
  float* hout = h + (size_t)(row0 + 8 * half) * HID + col;
#pragma unroll
  for (int r = 0; r < 8; ++r) hout[r * HID] = c[r];
}

// Layer-1 scatter: agg[dst,j] += h[src,j] * norm[e]; 16 lanes per edge.
__global__ void agg1_kernel(const long long* __restrict__ src,
                            const long long* __restrict__ dst,
                            const float* __restrict__ norm,
                            const float* __restrict__ h,
                            float* __restrict__ agg, int E) {
  long long idx = (long long)blockIdx.x * blockDim.x + threadIdx.x;
  long long e = idx >> 4;
  int j = (int)(idx & 15);
  if (e < E) {
    int s = (int)src[e];
    int d = (int)dst[e];
    float w = norm[e];
    atomicAdd(&agg[(size_t)d * HID + j], h[(size_t)s * HID + j] * w);
  }
}

// Per-node epilogue of layer 1 + dense part of layer 2:
//   out1 = relu(agg + h*selfnorm + b1);  h2 = out1 . W2
__global__ void post1(const float* __restrict__ agg,
                      const float* __restrict__ h,
                      const float* __restrict__ dis,
                      const float* __restrict__ b1,
                      const float* __restrict__ W2,
                      float* __restrict__ h2, int N) {
  int n = blockIdx.x * blockDim.x + threadIdx.x;
  if (n < N) {
    float sn = dis[n];
    sn *= sn;                            // self-loop norm = 1/deg
    float acc = 0.0f;
#pragma unroll
    for (int j = 0; j < HID; ++j) {
      float v = agg[(size_t)n * HID + j] + h[(size_t)n * HID + j] * sn + b1[j];
      v = fmaxf(v, 0.0f);
      acc += v * W2[j];
    }
    h2[n] = acc;
  }
}

// Layer-2 scatter: out[dst] += h2[src] * norm[e]
__global__ void agg2_kernel(const long long* __restrict__ src,
                            const long long* __restrict__ dst,
                            const float* __restrict__ norm,
                            const float* __restrict__ h2,
                            float* __restrict__ out, int E) {
  int e = blockIdx.x * blockDim.x + threadIdx.x;
  if (e < E) atomicAdd(&out[(int)dst[e]], h2[(int)src[e]] * norm[e]);
}

// out[n] += h2[n]*selfnorm + b2
__global__ void final_kernel(float* __restrict__ out,
                             const float* __restrict__ h2,
                             const float* __restrict__ dis,
                             const float* __restrict__ b2, int N) {
  int n = blockIdx.x * blockDim.x + threadIdx.x;
  if (n < N) {
    float s = dis[n];
    out[n] += h2[n] * s * s + b2[0];
  }
}

extern "C" void kernel_launch(void* const* d_in, const int* in_sizes, int n_in,
                              void* d_out, int out_size, void* d_ws,
                              size_t ws_size, hipStream_t stream) {
  const float*     x  = (const float*)d_in[0];
  const long long* ei = (const long long*)d_in[1];   // int64 [2, E]
  const float*     W1 = (const float*)d_in[2];       // [16, 128]
  const float*     b1 = (const float*)d_in[3];       // [16]
  const float*     W2 = (const float*)d_in[4];       // [1, 16]
  const float*     b2 = (const float*)d_in[5];       // [1]

  const int N = in_sizes[0] / IND;
  const int E = in_sizes[1] / 2;
  const long long* src = ei;
  const long long* dst = ei + E;
  float* out = (float*)d_out;

  // workspace layout (floats): deg/dis [N] | norm [E] | h [N*16] | agg [N*16] | h2 [N]
  float* ws   = (float*)d_ws;
  float* deg  = ws;                          // becomes dis in place
  float* norm = deg + N;
  float* h    = norm + E;
  float* agg  = h + (size_t)N * HID;
  float* h2   = agg + (size_t)N * HID;

  const int B = 256;
  // degrees + symmetric normalization (shared across both layers)
  zero_f32 <<<(N + B - 1) / B, B, 0, stream>>>(deg, N);
  deg_count<<<(E + B - 1) / B, B, 0, stream>>>(dst, E, deg);
  deg_to_dis<<<(N + B - 1) / B, B, 0, stream>>>(deg, N);
  edge_norm<<<(E + B - 1) / B, B, 0, stream>>>(src, dst, deg, norm, E);

  // layer 1 dense transform (WMMA) + scatter + epilogue (includes W2 dot)
  gemm1_wmma<<<(N / 16 + 7) / 8, 256, 0, stream>>>(x, W1, h, N);
  zero_f32  <<<((N * HID) + B - 1) / B, B, 0, stream>>>(agg, N * HID);
  long long tot = (long long)E * HID;
  agg1_kernel<<<(int)((tot + B - 1) / B), B, 0, stream>>>(src, dst, norm, h,
                                                          agg, E);
  post1<<<(N + B - 1) / B, B, 0, stream>>>(agg, h, deg, b1, W2, h2, N);

  // layer 2 scatter + final bias/self-loop
  zero_f32   <<<(N + B - 1) / B, B, 0, stream>>>(out, N);
  agg2_kernel<<<(E + B - 1) / B, B, 0, stream>>>(src, dst, norm, h2, out, E);
  final_kernel<<<(N + B - 1) / B, B, 0, stream>>>(out, h2, deg, b2, N);
}